// DeepPoolNet_78975858639084
// MI455X (gfx1250) — compile-verified
//
#include <hip/hip_runtime.h>
#include <hip/hip_bf16.h>
#include <math.h>

#define NNODES 50000
#define NEDGES 600000
#define HDIM   128
#define CDIM   10

typedef __attribute__((ext_vector_type(2))) float v2f;
typedef __attribute__((ext_vector_type(8))) float v8f;

__device__ __forceinline__ float elu1(float x) {
    return x > 0.0f ? x : expm1f(x);
}

// ---------------- degree / normalization ----------------

__global__ void __launch_bounds__(256) k_zero(float* __restrict__ p, int n) {
    int i = blockIdx.x * blockDim.x + threadIdx.x;
    if (i < n) p[i] = 0.0f;
}

__global__ void __launch_bounds__(256) k_deg(const int* __restrict__ dst,
                                             float* __restrict__ deg, int e) {
    int i = blockIdx.x * blockDim.x + threadIdx.x;
    if (i < e) unsafeAtomicAdd(&deg[dst[i]], 1.0f);
}

__global__ void __launch_bounds__(256) k_rsqrt(float* __restrict__ dinv, int n) {
    int i = blockIdx.x * blockDim.x + threadIdx.x;
    if (i < n) dinv[i] = rsqrtf(dinv[i] + 1.0f);  // +1 self-loop
}

// ---------------- WMMA f32 GEMM: y = (h @ W) * dinv[row]; acc = y ----------------
// Block = 256 threads = 8 wave32. Block owns 16 rows; wave w owns columns
// [16w, 16w+16). A-tile (16x128 f32, 8KB) staged in LDS; W streamed from L2.
// D = A(16x4) x B(4x16) + C, K=128 -> 32 x v_wmma_f32_16x16x4_f32.

__global__ void __launch_bounds__(256) k_gemm(const float* __restrict__ h,
                                              const float* __restrict__ W,
                                              const float* __restrict__ dinv,
                                              float* __restrict__ y,
                                              float* __restrict__ acc) {
    __shared__ float lA[16 * HDIM];  // 8 KB
    const int tid     = threadIdx.x;
    const int rowbase = blockIdx.x * 16;

    // cooperative float4 copy of the contiguous 16x128 tile
    const float4* hv = (const float4*)(h + (size_t)rowbase * HDIM);
    float4*       lv = (float4*)lA;
    lv[tid]       = hv[tid];
    lv[tid + 256] = hv[tid + 256];
    __syncthreads();

    const int wave  = tid >> 5;
    const int lane  = tid & 31;
    const int m     = lane & 15;             // row within tile / col within slice
    const int khalf = (lane >> 4) << 1;      // lanes 16-31 carry K+2,K+3
    const int n     = (wave << 4) + m;       // output column

    v8f c = {0.f, 0.f, 0.f, 0.f, 0.f, 0.f, 0.f, 0.f};
#pragma unroll
    for (int k = 0; k < HDIM; k += 4) {
        // A fragment: lane holds A[m][k+khalf], A[m][k+khalf+1]  (ds_load_b64)
        v2f a = *(const v2f*)&lA[m * HDIM + k + khalf];
        // B fragment: lane holds W[k+khalf][n], W[k+khalf+1][n]
        v2f b;
        b.x = W[(k + khalf) * HDIM + n];
        b.y = W[(k + khalf + 1) * HDIM + n];
        c = __builtin_amdgcn_wmma_f32_16x16x4_f32(false, a, false, b,
                                                  (short)0, c, false, false);
    }

    // C/D layout: VGPR i -> M=i (lanes 0-15), M=i+8 (lanes 16-31)
    const int rhalf = (lane >> 4) << 3;
#pragma unroll
    for (int i = 0; i < 8; ++i) {
        const int   row = rowbase + rhalf + i;
        const float val = c[i] * dinv[row];      // fold dinv[src] into message
        const int   idx = row * HDIM + n;
        y[idx]   = val;
        acc[idx] = val;                          // self-loop term pre-seeded
    }
}

// ---------------- edge scatter: acc[dst] += y[src] ----------------
// one wave32 per edge, float4 per lane -> 128 floats

__global__ void __launch_bounds__(256) k_scatter(const int* __restrict__ src,
                                                 const int* __restrict__ dst,
                                                 const float* __restrict__ y,
                                                 float* __restrict__ acc) {
    const int t    = blockIdx.x * blockDim.x + threadIdx.x;
    const int e    = t >> 5;
    const int lane = t & 31;
    if (e >= NEDGES) return;
    const int s = src[e];
    const int d = dst[e];
    const float4 v = *(const float4*)&y[(size_t)s * HDIM + lane * 4];
    float* p = &acc[(size_t)d * HDIM + lane * 4];
    unsafeAtomicAdd(p + 0, v.x);
    unsafeAtomicAdd(p + 1, v.y);
    unsafeAtomicAdd(p + 2, v.z);
    unsafeAtomicAdd(p + 3, v.w);
}

// ---------------- finalize: h_out = elu(acc * dinv[row] + b) ----------------

__global__ void __launch_bounds__(256) k_finalize(const float* __restrict__ acc,
                                                  const float* __restrict__ dinv,
                                                  const float* __restrict__ b,
                                                  float* __restrict__ hout) {
    const int i = blockIdx.x * blockDim.x + threadIdx.x;  // float4 index
    if (i >= NNODES * (HDIM / 4)) return;
    const int row  = i >> 5;          // 32 float4 per row
    const int col4 = (i & 31) * 4;
    const float  di = dinv[row];
    const float4 a  = *(const float4*)&acc[(size_t)row * HDIM + col4];
    const float4 bb = *(const float4*)&b[col4];
    float4 o;
    o.x = elu1(fmaf(a.x, di, bb.x));
    o.y = elu1(fmaf(a.y, di, bb.y));
    o.z = elu1(fmaf(a.z, di, bb.z));
    o.w = elu1(fmaf(a.w, di, bb.w));
    *(float4*)&hout[(size_t)row * HDIM + col4] = o;
}

// ---------------- final MLP (bug-faithful: only Wm2/bm2 survive) ----------------

__global__ void __launch_bounds__(256) k_mlp(const float* __restrict__ h,
                                             const float* __restrict__ Wm,
                                             const float* __restrict__ bm,
                                             float* __restrict__ S) {
    __shared__ float lw[HDIM * CDIM];  // 5 KB
    const int tid = threadIdx.x;
    for (int i = tid; i < HDIM * CDIM; i += blockDim.x) lw[i] = Wm[i];
    __syncthreads();
    const int node = blockIdx.x * blockDim.x + tid;
    if (node >= NNODES) return;
    float a[CDIM];
#pragma unroll
    for (int c = 0; c < CDIM; ++c) a[c] = bm[c];
    const float* hr = h + (size_t)node * HDIM;
    for (int k = 0; k < HDIM; ++k) {
        const float hv = hr[k];
#pragma unroll
        for (int c = 0; c < CDIM; ++c) a[c] = fmaf(hv, lw[k * CDIM + c], a[c]);
    }
    float* out = S + (size_t)node * CDIM;
#pragma unroll
    for (int c = 0; c < CDIM; ++c) out[c] = elu1(a[c]);
}

// ---------------- driver ----------------

extern "C" void kernel_launch(void* const* d_in, const int* in_sizes, int n_in,
                              void* d_out, int out_size, void* d_ws, size_t ws_size,
                              hipStream_t stream) {
    const float* x   = (const float*)d_in[0];
    const int*   ei  = (const int*)d_in[1];
    const int*   src = ei;
    const int*   dst = ei + NEDGES;
    const float* W[4] = {(const float*)d_in[2], (const float*)d_in[4],
                         (const float*)d_in[6], (const float*)d_in[8]};
    const float* b[4] = {(const float*)d_in[3], (const float*)d_in[5],
                         (const float*)d_in[7], (const float*)d_in[9]};
    const float* Wm2 = (const float*)d_in[12];
    const float* bm2 = (const float*)d_in[13];

    float* out_h = (float*)d_out;
    float* out_S = out_h + (size_t)NNODES * HDIM;

    float* ws   = (float*)d_ws;
    float* y    = ws;                               // N*128
    float* acc  = y + (size_t)NNODES * HDIM;        // N*128
    float* hbuf = acc + (size_t)NNODES * HDIM;      // N*128
    float* dinv = hbuf + (size_t)NNODES * HDIM;     // N

    // normalization (deg from dst, +1 self-loop, rsqrt) — recomputed each call
    k_zero<<<(NNODES + 255) / 256, 256, 0, stream>>>(dinv, NNODES);
    k_deg<<<(NEDGES + 255) / 256, 256, 0, stream>>>(dst, dinv, NEDGES);
    k_rsqrt<<<(NNODES + 255) / 256, 256, 0, stream>>>(dinv, NNODES);

    const float* hin = x;
    float* houts[4] = {hbuf, out_h, hbuf, out_h};   // final h lands in d_out
    for (int l = 0; l < 4; ++l) {
        k_gemm<<<NNODES / 16, 256, 0, stream>>>(hin, W[l], dinv, y, acc);
        k_scatter<<<(NEDGES * 32) / 256, 256, 0, stream>>>(src, dst, y, acc);
        k_finalize<<<(NNODES * 32) / 256, 256, 0, stream>>>(acc, dinv, b[l], houts[l]);
        hin = houts[l];
    }

    k_mlp<<<(NNODES + 255) / 256, 256, 0, stream>>>(out_h, Wm2, bm2, out_S);
}